// LiquidEchoHead_75892072121029
// MI455X (gfx1250) — compile-verified
//
#include <hip/hip_runtime.h>
#include <cstdint>
#include <math.h>

// ---------------------------------------------------------------------------
// LiquidEchoHead for MI455X (gfx1250).
// Bandwidth-bound elementwise + per-row reduction kernel.
// One workgroup (256 threads, 8 wave32 waves) handles ROWS=4 rows of D=2048.
// Row inputs are staged into LDS by the Tensor Data Mover (TENSORcnt) and the
// DMA of row r+1 is overlapped with the compute of row r (registers act as
// the double buffer). A shared sin LUT (cos = sin shifted 1024 entries) is
// built once per block with v_sin_f32 while the first DMA is in flight.
// ---------------------------------------------------------------------------

#define LUT_N    4096
#define LUT_EXT  (LUT_N + 1025)      // sin table + cos offset (+1024) + interp (+1)
#define D_MODEL  2048
#define BLOCK    256
#define EPT      (D_MODEL / BLOCK)   // 8 columns per thread
#define ROWS     4                   // rows per block (amortize LUT + params)
#define NW       (BLOCK / 32)        // waves per block

#define TWO_PI_F     6.2831853071795864769f
#define LUT_SCALE_F  651.8986469044033f       // LUT_N / TWO_PI
#define INV_LUT_N_F  (1.0f / 4096.0f)
#define PHI_F        1.6180339887498948f
#define INV_SQRT_D   0.02209708691207961f     // 1/sqrt(2048)

typedef unsigned int u32x4 __attribute__((ext_vector_type(4)));
typedef int          i32x4 __attribute__((ext_vector_type(4)));
typedef int          i32x8 __attribute__((ext_vector_type(8)));

#if __has_builtin(__builtin_amdgcn_tensor_load_to_lds)
#define HAVE_TDM 1
#else
#define HAVE_TDM 0
#endif

// DMA one contiguous row of n f32 elements from global into LDS via TDM.
// Descriptor packing per CDNA5 ISA §8 (D# group0 + group1; groups 2/3 NULL).
__device__ __forceinline__ void tdm_stage_row(const float* gsrc,
                                              const float* lds_dst,
                                              unsigned n) {
#if HAVE_TDM
  unsigned long long ga = (unsigned long long)(uintptr_t)gsrc;
  // Generic shared pointers carry the LDS byte offset in their low 32 bits.
  unsigned lds_off = (unsigned)(unsigned long long)(uintptr_t)lds_dst;

  u32x4 g0;
  g0[0] = 1u;                                            // count=1, user mode
  g0[1] = lds_off;                                       // lds_addr
  g0[2] = (unsigned)(ga & 0xFFFFFFFFull);                // global_addr[31:0]
  g0[3] = (unsigned)((ga >> 32) & 0x01FFFFFFull)         // global_addr[56:32]
        | (2u << 30);                                    // type = 2 ("image")

  i32x8 g1;
  g1[0] = 0x00020000;                                    // data_size=2 (4B), mask=0
  g1[1] = (int)((n & 0xFFFFu) << 16);                    // tensor_dim0[15:0]
  g1[2] = (int)(((n >> 16) & 0xFFFFu) | (1u << 16));     // dim0[31:16] | tensor_dim1=1
  g1[3] = (int)((n & 0xFFFFu) << 16);                    // tile_dim0 = n
  g1[4] = 1;                                             // tile_dim1=1, tile_dim2=0
  g1[5] = (int)n;                                        // tensor_dim0_stride[31:0]
  g1[6] = 0;                                             // stride[47:32], dim1_stride lo
  g1[7] = 0;

  i32x4 z4 = {0, 0, 0, 0};
#if defined(__clang_major__) && (__clang_major__ >= 23)
  i32x8 z8 = {0, 0, 0, 0, 0, 0, 0, 0};
  __builtin_amdgcn_tensor_load_to_lds(g0, g1, z4, z4, z8, 0);
#else
  __builtin_amdgcn_tensor_load_to_lds(g0, g1, z4, z4, 0);
#endif
#else
  // Fallback (no TDM builtin): cooperative copy by the calling wave.
  float* dst = const_cast<float*>(lds_dst);
  for (unsigned i = threadIdx.x & 31u; i < n; i += 32u) dst[i] = gsrc[i];
#endif
}

__device__ __forceinline__ void wait_tdm() {
#if HAVE_TDM
#if __has_builtin(__builtin_amdgcn_s_wait_tensorcnt)
  __builtin_amdgcn_s_wait_tensorcnt(0);
#else
  asm volatile("s_wait_tensorcnt 0" ::: "memory");
#endif
#endif
}

// Linearly interpolated LUT sin/cos (cos(x) = sin(x + pi/2) = T[i+1024]).
// Matches the reference's 4096-entry table semantics including wrap.
__device__ __forceinline__ void lut_sincos(const float* T, float theta,
                                           float& s, float& c) {
  float u    = theta * LUT_SCALE_F;
  float fl   = floorf(u * INV_LUT_N_F);
  float pos  = fmaf(-fl, 4096.0f, u);     // mod into ~[0, 4096)
  float i0f  = floorf(pos);
  float frac = pos - i0f;
  int   i0   = ((int)i0f) & (LUT_N - 1);  // handles the tiny-negative edge too
  float s0 = T[i0],        s1 = T[i0 + 1];
  float c0 = T[i0 + 1024], c1 = T[i0 + 1025];
  s = fmaf(s1 - s0, frac, s0);
  c = fmaf(c1 - c0, frac, c0);
}

__device__ __forceinline__ void load8(const float* p, int tid, float v[EPT]) {
  const float4* p4 = (const float4*)p;
  float4 a = p4[2 * tid], b = p4[2 * tid + 1];
  v[0] = a.x; v[1] = a.y; v[2] = a.z; v[3] = a.w;
  v[4] = b.x; v[5] = b.y; v[6] = b.z; v[7] = b.w;
}

__device__ __forceinline__ void store8(float* p, int tid, const float v[EPT]) {
  float4* p4 = (float4*)p;
  p4[2 * tid]     = make_float4(v[0], v[1], v[2], v[3]);
  p4[2 * tid + 1] = make_float4(v[4], v[5], v[6], v[7]);
}

__global__ __launch_bounds__(BLOCK) void liquid_echo_kernel(
    const float* __restrict__ x_real,  const float* __restrict__ x_imag,
    const float* __restrict__ t_vec,   const float* __restrict__ w_query,
    const float* __restrict__ b_query, const float* __restrict__ w_osc,
    const float* __restrict__ b_osc,   const float* __restrict__ mem_real,
    const float* __restrict__ mem_imag,
    float* __restrict__ out_real, float* __restrict__ out_imag, int B) {
  __shared__ __align__(16) float s_sin[LUT_EXT];
  __shared__ __align__(16) float s_row[4 * D_MODEL];  // xr | xi | mr | mi
  __shared__ float s_red[2 * NW];
  __shared__ float s_alpha;

  const int tid  = threadIdx.x;
  const int row0 = blockIdx.x * ROWS;

  // Kick off the TDM DMA for the first row (wave 0 issues; SGPR descriptors,
  // tracked by TENSORcnt).
  if (tid < 32) {
    const size_t b0 = (size_t)row0 * D_MODEL;
    tdm_stage_row(x_real   + b0, s_row + 0 * D_MODEL, D_MODEL);
    tdm_stage_row(x_imag   + b0, s_row + 1 * D_MODEL, D_MODEL);
    tdm_stage_row(mem_real + b0, s_row + 2 * D_MODEL, D_MODEL);
    tdm_stage_row(mem_imag + b0, s_row + 3 * D_MODEL, D_MODEL);
  }

  // Prefetch the L2-resident parameter vectors (global_prefetch_b8).
  __builtin_prefetch(w_query + tid * EPT, 0, 1);
  __builtin_prefetch(b_query + tid * EPT, 0, 1);
  __builtin_prefetch(w_osc   + tid * EPT, 0, 1);
  __builtin_prefetch(b_osc   + tid * EPT, 0, 1);

  // Build the extended sin LUT (once per block) while the DMA is in flight.
  for (int j = tid; j < LUT_EXT; j += BLOCK)
    s_sin[j] = __sinf((float)(j & (LUT_N - 1)) * (TWO_PI_F / (float)LUT_N));

  // Hoist row-invariant parameter math out of the row loop.
  float invwq[EPT], bq[EPT], invwo[EPT], bo[EPT];
  load8(w_query, tid, invwq);
  load8(b_query, tid, bq);
  load8(w_osc, tid, invwo);
  load8(b_osc, tid, bo);
#pragma unroll
  for (int i = 0; i < EPT; ++i) {
    invwq[i] = 1.0f / (1.0f + fabsf(invwq[i]));
    invwo[i] = 1.0f / (1.0f + fabsf(invwo[i]));
  }

  for (int r = 0; r < ROWS; ++r) {
    const int row = row0 + r;
    if (row >= B) break;
    const size_t base = (size_t)row * D_MODEL;

    // Current row's DMA was issued earlier; make it visible block-wide.
    if (tid < 32) wait_tdm();
    __syncthreads();

    // Drain the staging buffer into registers (registers = double buffer).
    float xr[EPT], xi[EPT], mr[EPT], mi[EPT];
    load8(s_row + 0 * D_MODEL, tid, xr);
    load8(s_row + 1 * D_MODEL, tid, xi);
    load8(s_row + 2 * D_MODEL, tid, mr);
    load8(s_row + 3 * D_MODEL, tid, mi);
    __syncthreads();  // everyone done reading: buffer is reusable

    // Overlap: launch the next row's DMA before computing this row.
    if (tid < 32 && (row + 1) < B && (r + 1) < ROWS) {
      const size_t bn = (size_t)(row + 1) * D_MODEL;
      tdm_stage_row(x_real   + bn, s_row + 0 * D_MODEL, D_MODEL);
      tdm_stage_row(x_imag   + bn, s_row + 1 * D_MODEL, D_MODEL);
      tdm_stage_row(mem_real + bn, s_row + 2 * D_MODEL, D_MODEL);
      tdm_stage_row(mem_imag + bn, s_row + 3 * D_MODEL, D_MODEL);
    }

    // -------------- Phase 1: query LUT + interference reduction -----------
    float accR = 0.0f, accI = 0.0f;
#pragma unroll
    for (int i = 0; i < EPT; ++i) {
      float th = fmaf(xr[i], invwq[i], bq[i]);
      float s, c;
      lut_sincos(s_sin, th, s, c);
      accR = fmaf(c, xr[i], fmaf(s, xi[i], accR));
      accI = fmaf(c, xi[i], fmaf(-s, xr[i], accI));
    }
    // wave32 tree reduction, then cross-wave combine in LDS
    for (int off = 16; off > 0; off >>= 1) {
      accR += __shfl_xor(accR, off, 32);
      accI += __shfl_xor(accI, off, 32);
    }
    const int wid = tid >> 5;
    if ((tid & 31) == 0) {
      s_red[wid]      = accR;
      s_red[NW + wid] = accI;
    }
    __syncthreads();
    if (tid == 0) {
      float rr = 0.0f, im = 0.0f;
      for (int w = 0; w < NW; ++w) {
        rr += s_red[w];
        im += s_red[NW + w];
      }
      float interf = sqrtf(fmaf(rr, rr, im * im));
      float sig    = 1.0f / (1.0f + __expf(-(interf * INV_SQRT_D - 2.0f)));
      s_alpha      = __expf(-(1.0f - sig));   // k = 1
    }
    __syncthreads();

    // -------------- Phase 2: blend + oscillator evolution ------------------
    const float alpha = s_alpha;
    const float oma   = 1.0f - alpha;
    const float tphi  = t_vec[row] * PHI_F;

    float er[EPT], ei[EPT];
#pragma unroll
    for (int i = 0; i < EPT; ++i) {
      float br = fmaf(alpha, xr[i], oma * mr[i]);
      float bi = fmaf(alpha, xi[i], oma * mi[i]);
      float tb = bo[i] + tphi;
      float sr, cr, si, ci;
      lut_sincos(s_sin, fmaf(br, invwo[i], tb), sr, cr);
      lut_sincos(s_sin, fmaf(bi, invwo[i], tb), si, ci);
      er[i] = fmaf(cr, ci, -(sr * si));
      ei[i] = fmaf(cr, si, sr * ci);
    }
    store8(out_real + base, tid, er);
    store8(out_imag + base, tid, ei);
  }
}

extern "C" void kernel_launch(void* const* d_in, const int* in_sizes, int n_in,
                              void* d_out, int out_size, void* d_ws,
                              size_t ws_size, hipStream_t stream) {
  (void)n_in; (void)out_size; (void)d_ws; (void)ws_size;
  const float* x_real   = (const float*)d_in[0];
  const float* x_imag   = (const float*)d_in[1];
  const float* t_vec    = (const float*)d_in[2];
  const float* w_query  = (const float*)d_in[3];
  const float* b_query  = (const float*)d_in[4];
  const float* w_osc    = (const float*)d_in[5];
  const float* b_osc    = (const float*)d_in[6];
  const float* mem_real = (const float*)d_in[7];
  const float* mem_imag = (const float*)d_in[8];

  const int B = in_sizes[2];  // t has B elements; D fixed at 2048 per reference
  float* out_real = (float*)d_out;
  float* out_imag = out_real + (size_t)B * D_MODEL;

  const int grid = (B + ROWS - 1) / ROWS;
  liquid_echo_kernel<<<dim3(grid), dim3(BLOCK), 0, stream>>>(
      x_real, x_imag, t_vec, w_query, b_query, w_osc, b_osc, mem_real,
      mem_imag, out_real, out_imag, B);
}